// DistanceTransformLoss_11158325035008
// MI455X (gfx1250) — compile-verified
//
#include <hip/hip_runtime.h>
#include <hip/hip_bf16.h>
#include <math.h>

#define Nn 8
#define Cc 19
#define Hh 512
#define Ww 512
#define HW (Hh*Ww)            // 262144 = 2^18
#define NPIX (Nn*HW)          // 2097152
#define PEN 1033              // N + 1 + H + W
#define DILATE 5
#define NBLK1 (NPIX/256)      // 8192
#define NCOL (Nn*Cc*Ww)       // 77824 = 304*256
#define NROW (Nn*Cc*Hh)       // 77824

typedef float v2f __attribute__((ext_vector_type(2)));
typedef float v8f __attribute__((ext_vector_type(8)));

// ---------------- init ----------------
__global__ __launch_bounds__(64) void k_init(unsigned int* pres) {
    if (threadIdx.x == 0 && blockIdx.x == 0) *pres = 0u;
}

// ------------- pass 1: per-pixel softmax stats, CE, argmax, present -------------
__global__ __launch_bounds__(256) void k_pixel(const float* __restrict__ slices,
                                               const int* __restrict__ targets,
                                               unsigned char* __restrict__ pred,
                                               float* __restrict__ maxv,
                                               float* __restrict__ rse,
                                               float* __restrict__ pce,
                                               unsigned int* __restrict__ pres) {
    int tid = threadIdx.x;
    int idx = blockIdx.x * 256 + tid;
    int n  = idx >> 18;
    int hw = idx & (HW - 1);
    size_t base = (((size_t)(n * Cc)) << 18) + hw;

    int t = targets[idx];
    float xs[Cc];
    float m = -3.4e38f; int am = 0; float xt = 0.f;
    #pragma unroll
    for (int c = 0; c < Cc; c++) {
        float v = slices[base + ((size_t)c << 18)];
        xs[c] = v;
        if (v > m) { m = v; am = c; }   // strict > keeps first max (argmax tie rule)
        if (c == t) xt = v;
    }
    float se = 0.f;
    #pragma unroll
    for (int c = 0; c < Cc; c++) se += expf(xs[c] - m);

    float ce = 0.f;
    unsigned pbit = 0u;
    if (t >= 0 && t < Cc) {             // t != IGNORE
        ce   = -(xt - m - logf(se));
        pbit = 1u << t;
    }
    pred[idx] = (unsigned char)am;
    maxv[idx] = m;
    rse[idx]  = 1.0f / se;

    __shared__ float    sred[256];
    __shared__ unsigned sob[256];
    sred[tid] = ce; sob[tid] = pbit;
    __syncthreads();
    for (int s = 128; s > 0; s >>= 1) {
        if (tid < s) { sred[tid] += sred[tid + s]; sob[tid] |= sob[tid + s]; }
        __syncthreads();
    }
    if (tid == 0) { pce[blockIdx.x] = sred[0]; atomicOr(pres, sob[0]); }
}

// ------------- pass 2a: vertical (H-axis) distance transform, fused tgt-boundary -------------
__global__ __launch_bounds__(256) void k_dtvert(const int* __restrict__ targets,
                                                unsigned short* __restrict__ dt) {
    int tid = blockIdx.x * 256 + threadIdx.x;      // exactly NCOL threads
    int w = tid & (Ww - 1);
    int c = (tid >> 9) % Cc;
    int n = tid / (Ww * Cc);
    const int* tg = targets + ((size_t)n << 18);
    unsigned short* col = dt + (((size_t)(n * Cc + c)) << 18);

    int carry = 1 << 20;
    int tcur = tg[w];                               // h = 0
    for (int h = 0; h < Hh; h++) {
        int o = (h << 9) + w;
        int td = (h < Hh - 1) ? tg[o + Ww] : -1;
        bool b = false;
        if (h < Hh - 1 && tcur != td && (c == tcur || c == td)) b = true;
        if (w > 0) { int tl = tg[o - 1]; if (tcur != tl && (c == tcur || c == tl)) b = true; }
        int x = b ? 0 : PEN;
        int cp1 = carry + 1;
        int f = x < cp1 ? x : cp1;
        carry = f;
        col[o] = (unsigned short)f;
        tcur = td;
    }
    carry = 1 << 20;
    for (int h = Hh - 1; h >= 0; h--) {
        int o = (h << 9) + w;
        int v = col[o];
        int cp1 = carry + 1;
        int r = v < cp1 ? v : cp1;
        carry = r;
        col[o] = (unsigned short)r;
    }
}

// ------------- pass 2b: horizontal (W-axis) DT as parallel prefix/suffix min, + relu(d-5) -------------
__global__ __launch_bounds__(512) void k_dthoriz(unsigned short* __restrict__ dt) {
    int row = blockIdx.x;                           // (n*C+c)*H + h
    int w = threadIdx.x;
    unsigned short* line = dt + (((size_t)(row >> 9)) << 18) + ((size_t)(row & (Hh - 1)) << 9);

    __shared__ int sa[Ww], sb[Ww];
    int x = line[w];
    sa[w] = x - w;                                  // prefix-min domain
    sb[w] = x + w;                                  // suffix-min domain
    __syncthreads();
    for (int off = 1; off < Ww; off <<= 1) {
        int va = sa[w], vb = sb[w];
        if (w >= off)       { int u = sa[w - off]; va = u < va ? u : va; }
        if (w + off < Ww)   { int u = sb[w + off]; vb = u < vb ? u : vb; }
        __syncthreads();
        sa[w] = va; sb[w] = vb;
        __syncthreads();
    }
    int e1 = w + sa[w];
    int e2 = sb[w] - w;
    int e = e1 < e2 ? e1 : e2;                      // min_j (x_j + |w-j|)
    e -= DILATE; if (e < 0) e = 0;                  // relu(d - BORDER_DILATE)
    line[w] = (unsigned short)e;
}

// ------------- pass 3: border sum (<=3 boundary classes per pixel) -------------
__global__ __launch_bounds__(256) void k_border(const float* __restrict__ slices,
                                                const unsigned char* __restrict__ pred,
                                                const float* __restrict__ maxv,
                                                const float* __restrict__ rse,
                                                const unsigned short* __restrict__ dt,
                                                const unsigned int* __restrict__ pres,
                                                float* __restrict__ pbd) {
    int tid = threadIdx.x;
    int idx = blockIdx.x * 256 + tid;
    int n  = idx >> 18;
    int hw = idx & (HW - 1);
    int h = hw >> 9;
    int w = hw & (Ww - 1);

    int p = pred[idx];
    unsigned bits = 0u;
    if (h < Hh - 1) { int pd = pred[idx + Ww]; if (pd != p) bits |= (1u << p) | (1u << pd); }
    if (w > 0)      { int pl = pred[idx - 1];  if (pl != p) bits |= (1u << p) | (1u << pl); }
    bits &= *pres;                                  // present[c] gate

    float acc = 0.f;
    if (bits) {
        float m = maxv[idx], r = rse[idx];
        size_t base = (((size_t)(n * Cc)) << 18) + hw;
        do {
            int c = __ffs(bits) - 1; bits &= bits - 1;
            size_t o = base + ((size_t)c << 18);
            float sm = expf(slices[o] - m) * r;
            acc += sm * (float)dt[o];
        } while (bits);
    }
    __shared__ float sred[256];
    sred[tid] = acc;
    __syncthreads();
    for (int s = 128; s > 0; s >>= 1) {
        if (tid < s) sred[tid] += sred[tid + s];
        __syncthreads();
    }
    if (tid == 0) pbd[blockIdx.x] = sred[0];
}

// ------------- pass 4: deterministic final reduction via V_WMMA_F32_16X16X4_F32 -------------
// D = A x B + C with B = all-ones: each C element accumulates a row-sum of A.
// One wave (32 lanes) consumes 64 partials per WMMA; sum over rows recovered via shfl.
__global__ __launch_bounds__(32) void k_final(const float* __restrict__ pce,
                                              const float* __restrict__ pbd,
                                              float* __restrict__ out) {
    int lane = threadIdx.x;
    v8f ac = {0.f,0.f,0.f,0.f,0.f,0.f,0.f,0.f};
    v8f ab = {0.f,0.f,0.f,0.f,0.f,0.f,0.f,0.f};
    v2f ones; ones.x = 1.f; ones.y = 1.f;
    for (int i = 0; i < NBLK1; i += 64) {
        v2f a; a.x = pce[i + 2*lane]; a.y = pce[i + 2*lane + 1];
        ac = __builtin_amdgcn_wmma_f32_16x16x4_f32(false, a, false, ones, (short)0, ac, false, false);
        v2f b; b.x = pbd[i + 2*lane]; b.y = pbd[i + 2*lane + 1];
        ab = __builtin_amdgcn_wmma_f32_16x16x4_f32(false, b, false, ones, (short)0, ab, false, false);
    }
    float sc = 0.f, sb = 0.f;
    #pragma unroll
    for (int k = 0; k < 8; k++) { sc += ac[k]; sb += ab[k]; }
    // lanes 0..15 hold rows 0..7 of column n=lane; lanes 16..31 hold rows 8..15.
    float ceT = __shfl(sc, 0) + __shfl(sc, 16);
    float bdT = __shfl(sb, 0) + __shfl(sb, 16);
    if (lane == 0) {
        if (bdT < 0.f) bdT = 0.f;
        out[0] = ceT + sqrtf(bdT);                  // BORDER_WEIGHT=1, power=0.5
    }
}

extern "C" void kernel_launch(void* const* d_in, const int* in_sizes, int n_in,
                              void* d_out, int out_size, void* d_ws, size_t ws_size,
                              hipStream_t stream) {
    const float* slices = (const float*)d_in[0];
    const int*  targets = (const int*)d_in[1];
    float* out = (float*)d_out;

    char* w = (char*)d_ws;
    // workspace layout (bytes):
    unsigned char*  pred = (unsigned char*)w;                               // 2 MB (padded to 4 MB)
    float*          maxv = (float*)(w + (size_t)4  * 1024 * 1024);          // 8 MB
    float*          rse  = (float*)(w + (size_t)12 * 1024 * 1024);          // 8 MB
    unsigned short* dt   = (unsigned short*)(w + (size_t)20 * 1024 * 1024); // NPIX*C*2 = 76 MB
    size_t dtoff = (size_t)20 * 1024 * 1024 + (size_t)NPIX * Cc * 2;        // 100,663,296
    float*        pce  = (float*)(w + dtoff);                               // 32 KB
    float*        pbd  = (float*)(w + dtoff + NBLK1 * 4);                   // 32 KB
    unsigned int* pres = (unsigned int*)(w + dtoff + 2 * NBLK1 * 4);        // 4 B

    k_init  <<<1, 64, 0, stream>>>(pres);
    k_pixel <<<NBLK1, 256, 0, stream>>>(slices, targets, pred, maxv, rse, pce, pres);
    k_dtvert<<<NCOL / 256, 256, 0, stream>>>(targets, dt);
    k_dthoriz<<<NROW, 512, 0, stream>>>(dt);
    k_border<<<NBLK1, 256, 0, stream>>>(slices, pred, maxv, rse, dt, pres, pbd);
    k_final <<<1, 32, 0, stream>>>(pce, pbd, out);
}